// Model_72292889526944
// MI455X (gfx1250) — compile-verified
//
#include <hip/hip_runtime.h>
#include <math.h>

// Problem constants (from reference setup_inputs)
#define KK   8
#define BB   32
#define CC   3
#define HWSZ 36864                 // 192*192
#define CHW  (CC*HWSZ)             // 110592
#define BHW  (BB*HWSZ)             // 1179648
#define BCHW (BB*CC*HWSZ)          // 3538944
#define KB   (KK*BB)               // 256
#define OUT_FLOATS ((KK+1)*BCHW)   // 31850496 floats, then KK*BB int32 indices

#define GINVVAR (1.0f/(0.3f*0.3f))
#define EPSF    1e-5f

// workspace layout (float offsets)
#define WS_DIFFS  0
#define WS_MASKS  ((size_t)KK*BHW)             // 9437184
#define WS_VISD   (2u*(size_t)KK*BHW)          // 18874368
#define WS_VISA   (WS_VISD + KB)
#define WS_VISM   (WS_VISA + KB)
#define WS_COEFS  (WS_VISM + KB)
#define WS_IDXB   (WS_COEFS + KB)              // 32 int32 (current step)
#define WS_IDXALL (WS_IDXB + BB)               // 256 int32 (all steps)

typedef __attribute__((ext_vector_type(2))) float v2f;
typedef __attribute__((ext_vector_type(4))) float v4f;
typedef __attribute__((ext_vector_type(8))) float v8f;

// ---------------------------------------------------------------------------
// Kernel 1: fused gamma compositing + reconstruction + diffs_sq + masks copy.
// float4 per thread (HW divisible by 4). grid: BHW/4/256 blocks, exact.
// ---------------------------------------------------------------------------
__global__ void k_fused_recon(const float* __restrict__ images,
                              const float* __restrict__ apc,
                              const float* __restrict__ shp,
                              const float* __restrict__ zeta,
                              const float* __restrict__ back,
                              float* __restrict__ out,
                              float* __restrict__ ws) {
  const int t    = blockIdx.x * blockDim.x + threadIdx.x;   // < BHW/4
  const int p4   = t * 4;
  const int b    = p4 / HWSZ;        // 4-groups never straddle b (HW%4==0)
  const int pix  = p4 - b * HWSZ;

  const v4f i0 = *(const v4f*)(images + (size_t)b*CHW + 0*HWSZ + pix);
  const v4f i1 = *(const v4f*)(images + (size_t)b*CHW + 1*HWSZ + pix);
  const v4f i2 = *(const v4f*)(images + (size_t)b*CHW + 2*HWSZ + pix);

  float* __restrict__ diffs = ws + WS_DIFFS;
  float* __restrict__ masks = ws + WS_MASKS;

  v4f run = (v4f)(1.0f);
  v4f r0  = (v4f)(0.0f), r1 = (v4f)(0.0f), r2 = (v4f)(0.0f);

#pragma unroll
  for (int k = 0; k < KK; ++k) {
    const int kb = k*BB + b;
    const v4f s = *(const v4f*)(shp + (size_t)kb*HWSZ + pix);
    const float z = zeta[kb];
    const v4f x = s * z;
    const v4f g = x * run;
    run *= ((v4f)(1.0f) - x);

    const float* ap = apc + (size_t)kb*CHW + pix;
    const v4f a0 = *(const v4f*)(ap);
    const v4f a1 = *(const v4f*)(ap + HWSZ);
    const v4f a2 = *(const v4f*)(ap + 2*HWSZ);
    r0 = g*a0 + r0;
    r1 = g*a1 + r1;
    r2 = g*a2 + r2;

    const v4f d0 = a0 - i0, d1 = a1 - i1, d2 = a2 - i2;
    *(v4f*)(diffs + (size_t)kb*HWSZ + pix) = d0*d0 + d1*d1 + d2*d2;
    *(v4f*)(masks + (size_t)kb*HWSZ + pix) = s;
  }
  // background layer
  r0 = run * (*(const v4f*)(back + (size_t)b*CHW + 0*HWSZ + pix)) + r0;
  r1 = run * (*(const v4f*)(back + (size_t)b*CHW + 1*HWSZ + pix)) + r1;
  r2 = run * (*(const v4f*)(back + (size_t)b*CHW + 2*HWSZ + pix)) + r2;

  *(v4f*)(out + (size_t)b*CHW + 0*HWSZ + pix) = r0;   // out[0] = recon
  *(v4f*)(out + (size_t)b*CHW + 1*HWSZ + pix) = r1;
  *(v4f*)(out + (size_t)b*CHW + 2*HWSZ + pix) = r2;
}

// ---------------------------------------------------------------------------
// Kernel 2: per-(k,b) reductions via V_WMMA_F32_16X16X4_F32 row-sums.
// One block (8 waves) per (k,b). b128 loads; dual accumulators so the
// back-to-back WMMAs pipeline instead of chaining on one C/D register.
// ---------------------------------------------------------------------------
static __device__ inline v8f wmma_sum_acc(v2f a, v2f ones, v8f c) {
  // 8 args: (neg_a, A, neg_b, B, c_mod, C, reuse_a, reuse_b)
  return __builtin_amdgcn_wmma_f32_16x16x4_f32(
      false, a, false, ones, (short)0, c, false, false);
}

__global__ void k_reduce_vis(const float* __restrict__ ws_base,
                             float* __restrict__ visD,
                             float* __restrict__ visA,
                             float* __restrict__ visM) {
  const int kb   = blockIdx.x;               // 0..255
  const int wave = threadIdx.x >> 5;
  const int lane = threadIdx.x & 31;

  const float* __restrict__ m = ws_base + WS_MASKS + (size_t)kb*HWSZ;
  const float* __restrict__ d = ws_base + WS_DIFFS + (size_t)kb*HWSZ;

  v8f accM0 = {}, accM1 = {};   // sum(masks)
  v8f accD0 = {}, accD1 = {};   // sum(masks*diffs)
  v2f onesv; onesv.x = 1.0f; onesv.y = 1.0f;
  float mx = 0.0f;              // masks are always >= 0

  // HWSZ/128 = 288 chunks of 128 floats; 8 waves -> 36 chunks each
  for (int i = wave; i < HWSZ/128; i += 8) {
    const int base = i*128 + lane*4;
    const v4f mv = *(const v4f*)(m + base);
    const v4f dv = *(const v4f*)(d + base);
    const v4f md = mv * dv;
    v2f mlo; mlo.x = mv.x;  mlo.y = mv.y;
    v2f mhi; mhi.x = mv.z;  mhi.y = mv.w;
    v2f dlo; dlo.x = md.x;  dlo.y = md.y;
    v2f dhi; dhi.x = md.z;  dhi.y = md.w;
    accM0 = wmma_sum_acc(mlo, onesv, accM0);
    accM1 = wmma_sum_acc(mhi, onesv, accM1);
    accD0 = wmma_sum_acc(dlo, onesv, accD0);
    accD1 = wmma_sum_acc(dhi, onesv, accD1);
    mx = fmaxf(mx, fmaxf(fmaxf(mv.x, mv.y), fmaxf(mv.z, mv.w)));
  }

  // fold D matrices: each lane sums its 8 rows; halves (M 0-7 / 8-15) via xor16
  float sM = 0.0f, sD = 0.0f;
#pragma unroll
  for (int j = 0; j < 8; ++j) {
    sM += accM0[j] + accM1[j];
    sD += accD0[j] + accD1[j];
  }
  sM += __shfl_xor(sM, 16, 32);
  sD += __shfl_xor(sD, 16, 32);
#pragma unroll
  for (int off = 16; off > 0; off >>= 1)
    mx = fmaxf(mx, __shfl_xor(mx, off, 32));

  __shared__ float lD[8], lA[8], lM[8];
  if (lane == 0) { lD[wave] = sD; lA[wave] = sM; lM[wave] = mx; }
  __syncthreads();
  if (threadIdx.x == 0) {
    float tD = 0.0f, tA = 0.0f, tM = 0.0f;
#pragma unroll
    for (int w = 0; w < 8; ++w) {
      tD += lD[w]; tA += lA[w]; tM = fmaxf(tM, lM[w]);
    }
    visD[kb] = tD; visA[kb] = tA; visM[kb] = tM;
  }
}

// ---------------------------------------------------------------------------
// Kernel 3: per-batch score + argmax over K; update coefs; record index.
// 1 block, 32 threads (one per b). Re-inits coefs at step 0 (replay-safe).
// ---------------------------------------------------------------------------
__global__ void k_select(const float* __restrict__ visD,
                         const float* __restrict__ visA,
                         const float* __restrict__ visM,
                         const float* __restrict__ zeta,
                         float* __restrict__ coefs,
                         int* __restrict__ idxB,
                         int* __restrict__ idxAll,
                         int* __restrict__ outIdx,
                         int step) {
  const int b = threadIdx.x;
  if (b >= BB) return;
  if (step == 0) {
#pragma unroll
    for (int k = 0; k < KK; ++k) coefs[k*BB + b] = 1.0f;
  }
  float best = -INFINITY;
  int bi = 0;
#pragma unroll
  for (int k = 0; k < KK; ++k) {
    const int kb = k*BB + b;
    const float sc = coefs[kb] * visM[kb] * zeta[kb] *
                     expf(-0.5f * GINVVAR * visD[kb] / (visA[kb] + EPSF));
    if (sc > best) { best = sc; bi = k; }   // first-occurrence argmax
  }
  idxB[b] = bi;
  idxAll[step*BB + b] = bi;
  coefs[bi*BB + b] = -1.0f;
  outIdx[step*BB + b] = bi;                 // indices output (int32 bits)
}

// ---------------------------------------------------------------------------
// Kernel 4: suppression: masks[k] *= (1 - masks[idx[b]]), sel read pre-update.
// float4 per thread. grid: BHW/4/256 blocks, exact.
// ---------------------------------------------------------------------------
__global__ void k_suppress(float* __restrict__ masks,
                           const int* __restrict__ idxB) {
  const int t   = blockIdx.x * blockDim.x + threadIdx.x;  // < BHW/4
  const int p4  = t * 4;
  const int b   = p4 / HWSZ;
  const int pix = p4 - b * HWSZ;
  const int id  = idxB[b];
  const v4f sel = *(const v4f*)(masks + (size_t)(id*BB + b)*HWSZ + pix);
  const v4f om  = (v4f)(1.0f) - sel;
#pragma unroll
  for (int k = 0; k < KK; ++k) {
    v4f* p = (v4f*)(masks + (size_t)(k*BB + b)*HWSZ + pix);
    *p = *p * om;
  }
}

// ---------------------------------------------------------------------------
// Kernel 5: reorder gather: out[1+k] = apc[indices[k]]. float4 per thread.
// ---------------------------------------------------------------------------
__global__ void k_gather(const float* __restrict__ apc,
                         const int* __restrict__ idxAll,
                         float* __restrict__ out) {
  const int t  = blockIdx.x * blockDim.x + threadIdx.x;   // < KK*BCHW/4
  const int e4 = t * 4;
  const int k  = e4 / BCHW;
  const int r  = e4 - k * BCHW;
  const int b  = r / CHW;
  const int id = idxAll[k*BB + b];
  *(v4f*)(out + (size_t)BCHW + e4) =
      *(const v4f*)(apc + (size_t)(id*BB + b)*CHW + (r - b*CHW));
}

// ---------------------------------------------------------------------------
extern "C" void kernel_launch(void* const* d_in, const int* in_sizes, int n_in,
                              void* d_out, int out_size, void* d_ws, size_t ws_size,
                              hipStream_t stream) {
  (void)in_sizes; (void)n_in; (void)out_size; (void)ws_size;
  const float* images = (const float*)d_in[0];
  const float* apc    = (const float*)d_in[1];
  const float* shp    = (const float*)d_in[2];
  const float* zeta   = (const float*)d_in[3];
  const float* back   = (const float*)d_in[4];
  float* out = (float*)d_out;
  float* ws  = (float*)d_ws;

  float* visD  = ws + WS_VISD;
  float* visA  = ws + WS_VISA;
  float* visM  = ws + WS_VISM;
  float* coefs = ws + WS_COEFS;
  int*   idxB  = (int*)(ws + WS_IDXB);
  int*   idxAll= (int*)(ws + WS_IDXALL);
  float* masks = ws + WS_MASKS;
  int*   outIdx= (int*)out + OUT_FLOATS;

  // 1) gamma/recon + diffs_sq + masks copy (out[0] written)
  k_fused_recon<<<(BHW/4)/256, 256, 0, stream>>>(images, apc, shp, zeta, back,
                                                 out, ws);

  // 2) greedy ordering loop (K steps)
  for (int step = 0; step < KK; ++step) {
    k_reduce_vis<<<KB, 256, 0, stream>>>(ws, visD, visA, visM);
    k_select<<<1, 32, 0, stream>>>(visD, visA, visM, zeta, coefs,
                                   idxB, idxAll, outIdx, step);
    k_suppress<<<(BHW/4)/256, 256, 0, stream>>>(masks, idxB);
  }

  // 3) apc reorder by indices (out[1..K])
  k_gather<<<(KK*BCHW/4)/256, 256, 0, stream>>>(apc, idxAll, out);
}